// BeamHypothesesModule_22634477650193
// MI455X (gfx1250) — compile-verified
//
#include <hip/hip_runtime.h>
#include <hip/hip_bf16.h>
#include <cstdint>

// Beam-hypotheses scoreboard update for MI455X (gfx1250, wave32).
// Bandwidth-bound permute-copy: items traffic ~604 MB -> use the CDNA5
// async global<->LDS data mover (ASYNCcnt) for the bulk copy, with the
// row permutation / hyp substitution folded into per-lane source addresses.
// Streams are read-once/write-once, so global traffic uses non-temporal
// cache hints to avoid sweeping L2.

namespace {

constexpr int NB = 9;             // NUM_BEAMS + 1 (scoreboard slots)
constexpr int PASS_CHUNKS = 2304; // int4 chunks staged per pass -> 36 KB LDS
constexpr int BLOCK = 256;        // 8 waves of 32

__global__ __launch_bounds__(BLOCK)
void beam_update_kernel(const int* __restrict__ hyp,          // (B, HW) int32
                        const float* __restrict__ slp_p,      // (B,) f32
                        const int* __restrict__ len_p,        // (B,) int32
                        const float* __restrict__ worst_p,    // (B,) f32
                        const float* __restrict__ board_p,    // (B, NB) f32
                        const int* __restrict__ items_p,      // (B, NB, W) int32
                        const int* __restrict__ curlen_p,     // scalar int32
                        int* __restrict__ out,                // len|worst|board|items
                        int batch, int w, int hw)
{
    __shared__ __align__(16) char smem[PASS_CHUNKS * 16];

    const int b   = (int)blockIdx.x;
    const int tid = (int)threadIdx.x;

    // ---- output section pointers (flat, in reference return order) ----
    int*   out_len   = out;
    float* out_worst = reinterpret_cast<float*>(out + batch);
    float* out_board = reinterpret_cast<float*>(out + 2 * (size_t)batch);
    int*   out_items = out + 2 * (size_t)batch + (size_t)batch * NB;

    // ---- per-row scalar logic (redundant per thread; all in registers) ----
    const float slp   = slp_p[b];
    const int   ln    = len_p[b];
    const float worst = worst_p[b];
    const int   c     = curlen_p[0];

    const float score   = slp / (float)c;           // LENGTH_PENALTY == 1.0
    const bool  cond    = (ln < NB - 1) || (score > worst);
    const int   new_len = ln + 1;
    const int   sel     = (new_len > NB) ? 0 : (NB - 1);

    float bd[NB];
    float my_old = 0.0f;                            // board[tid] for tid < NB
#pragma unroll
    for (int k = 0; k < NB; ++k) {
        bd[k] = board_p[(size_t)b * NB + k];
        if (k == tid) my_old = bd[k];
    }

    float b2[NB];
#pragma unroll
    for (int k = 0; k < NB; ++k) b2[k] = (k == sel) ? score : bd[k];

    // Stable ascending argsort of 9 floats via rank counting.
    // packed holds order[]: 4 bits per output slot.
    unsigned long long packed = 0ull;
    float b3_1 = b2[0];   // sorted[1]
    float myb3 = 0.0f;    // sorted[tid] for tid < NB
#pragma unroll
    for (int i = 0; i < NB; ++i) {
        int r = 0;
#pragma unroll
        for (int j = 0; j < NB; ++j) {
            const bool lt = (b2[j] < b2[i]) || ((b2[j] == b2[i]) && (j < i));
            r += lt ? 1 : 0;
        }
        packed |= (unsigned long long)i << (4 * r);
        if (r == 1)   b3_1 = b2[i];
        if (r == tid) myb3 = b2[i];
    }

    const float new_worst = (new_len > NB - 1) ? b3_1 : fminf(worst, score);

    if (tid == 0) {
        out_len[b]   = cond ? new_len : ln;
        out_worst[b] = cond ? new_worst : worst;
    }
    if (tid < NB) {
        out_board[(size_t)b * NB + tid] = cond ? myb3 : my_old;
    }

    // ---- bulk items permute-copy via async global<->LDS (ASYNCcnt) ----
    const int    w4        = w >> 2;                 // int4 chunks per row
    const int    total4    = NB * w4;                // chunks per batch row
    const size_t itemsBase = (size_t)b * NB * (size_t)w;
    // Flat address of LDS truncated to 32 bits == wave-relative LDS byte
    // offset (ISA 10.2 aperture mapping); also keeps smem allocated.
    const unsigned smem_base =
        (unsigned)(unsigned long long)(uintptr_t)&smem[0];

    for (int base = 0; base < total4; base += PASS_CHUNKS) {
        int lim = base + PASS_CHUNKS;
        if (lim > total4) lim = total4;

        // Issue gather loads: global -> LDS, 16 B per lane per issue.
        // Read-once stream -> non-temporal load hint.
        for (int v = base + tid; v < lim; v += BLOCK) {
            const int row  = v / w4;
            const int col4 = v - row * w4;
            int srow = row;
            if (cond) srow = (int)((packed >> (4 * row)) & 0xF);

            const int* gsrc;
            if (cond && (srow == sel) && ((col4 * 4 + 4) <= c)) {
                gsrc = hyp + (size_t)b * hw + (size_t)col4 * 4;      // hyp substitution
            } else {
                gsrc = items_p + itemsBase + (size_t)srow * w + (size_t)col4 * 4;
            }
            const unsigned lds = smem_base + (unsigned)(v - base) * 16u;
            const unsigned long long ga = (unsigned long long)(uintptr_t)gsrc;
            asm volatile("global_load_async_to_lds_b128 %0, %1, off th:TH_LOAD_NT"
                         :: "v"(lds), "v"(ga) : "memory");
        }
        asm volatile("s_wait_asynccnt 0x0" ::: "memory");

        // Issue streaming stores: LDS -> global.
        // Write-once, never re-read -> non-temporal store hint.
        for (int v = base + tid; v < lim; v += BLOCK) {
            const int row  = v / w4;
            const int col4 = v - row * w4;
            int* gdst = out_items + itemsBase + (size_t)row * w + (size_t)col4 * 4;
            const unsigned lds = smem_base + (unsigned)(v - base) * 16u;
            const unsigned long long ga = (unsigned long long)(uintptr_t)gdst;
            asm volatile("global_store_async_from_lds_b128 %0, %1, off th:TH_STORE_NT"
                         :: "v"(ga), "v"(lds) : "memory");
        }
        // WAR guard: next pass reuses this LDS staging region.
        asm volatile("s_wait_asynccnt 0x0" ::: "memory");
    }
}

} // namespace

extern "C" void kernel_launch(void* const* d_in, const int* in_sizes, int n_in,
                              void* d_out, int out_size, void* d_ws, size_t ws_size,
                              hipStream_t stream) {
    (void)n_in; (void)out_size; (void)d_ws; (void)ws_size;

    const int*   hyp    = (const int*)d_in[0];
    const float* slp    = (const float*)d_in[1];
    const int*   lenst  = (const int*)d_in[2];
    const float* worst  = (const float*)d_in[3];
    const float* board  = (const float*)d_in[4];
    const int*   items  = (const int*)d_in[5];
    const int*   curlen = (const int*)d_in[6];

    const int batch = in_sizes[1];                 // (BATCH,) sum_logprobs
    const int w     = in_sizes[5] / in_sizes[4];   // MAX_GEN_LEN (2048)
    const int hw    = in_sizes[0] / batch;         // CUR_LEN width of hyp (128)

    beam_update_kernel<<<dim3((unsigned)batch), dim3(BLOCK), 0, stream>>>(
        hyp, slp, lenst, worst, board, items, curlen,
        (int*)d_out, batch, w, hw);
}